// SelectiveSSM_83537113907375
// MI455X (gfx1250) — compile-verified
//
#include <hip/hip_runtime.h>
#include <math.h>

// ---- problem constants (match reference) ----
#define B_   2
#define L_   2048
#define D_   1024
#define N_   16
#define R_   64
#define E_   96          // R + 2N
#define CHUNKS 32
#define TCH  (L_ / CHUNKS)   // 64 steps per chunk

typedef float v2f __attribute__((ext_vector_type(2)));
typedef float v8f __attribute__((ext_vector_type(8)));

// ---------------------------------------------------------------------------
// Kernel 1: x_dbl[m][e] = sum_k x[m][k] * Wx[e][k]
//   M = B*L = 4096 (16-row tiles: 256), E = 96 (16-col tiles: 6), K = 1024.
//   One wave per 16x16 output tile, K-loop of 4 via V_WMMA_F32_16X16X4_F32.
// ---------------------------------------------------------------------------
__global__ __launch_bounds__(128) void k_xproj(const float* __restrict__ x,
                                               const float* __restrict__ Wx,
                                               float* __restrict__ xdbl) {
  const int wave = threadIdx.x >> 5;
  const int lane = threadIdx.x & 31;
  const int tile = blockIdx.x * 4 + wave;       // 0..1535
  const int mtile = tile / 6;                   // 0..255
  const int etile = tile % 6;                   // 0..5
  const int low = lane & 15;                    // A: row M, B/D: col N
  const int hi  = lane >> 4;                    // K-half / M-half select

  const float* Arow = x  + (size_t)(mtile * 16 + low) * D_;   // A[M=low][k]
  const float* Brow = Wx + (size_t)(etile * 16 + low) * D_;   // B[k][N=low] = Wx[N][k]

  v8f acc = {};
  for (int k = 0; k < D_; k += 4) {
    v2f a = *(const v2f*)(Arow + k + 2 * hi);   // K = k + 2*hi + {0,1}
    v2f b = *(const v2f*)(Brow + k + 2 * hi);
    acc = __builtin_amdgcn_wmma_f32_16x16x4_f32(false, a, false, b,
                                                (short)0, acc, false, false);
  }
  // D layout: VGPR r -> row (r + 8*hi), col = low
  float* Out = xdbl + (size_t)(mtile * 16) * E_ + etile * 16 + low;
#pragma unroll
  for (int r = 0; r < 8; ++r)
    Out[(size_t)(r + 8 * hi) * E_] = acc[r];
}

// ---------------------------------------------------------------------------
// Kernel 2: dt[m][d] = softplus( sum_r xdbl[m][r] * Wdt[d][r] + bdt[d] )
//   M = 4096 (256 tiles), Dcols = 1024 (64 tiles), K = 64.
// ---------------------------------------------------------------------------
__global__ __launch_bounds__(128) void k_dtproj(const float* __restrict__ xdbl,
                                                const float* __restrict__ Wdt,
                                                const float* __restrict__ bdt,
                                                float* __restrict__ dt) {
  const int wave = threadIdx.x >> 5;
  const int lane = threadIdx.x & 31;
  const int tile = blockIdx.x * 4 + wave;       // 0..16383
  const int mtile = tile >> 6;                  // 0..255
  const int dtile = tile & 63;                  // 0..63
  const int low = lane & 15;
  const int hi  = lane >> 4;

  const float* Arow = xdbl + (size_t)(mtile * 16 + low) * E_;   // dt_raw cols 0..63
  const float* Brow = Wdt  + (size_t)(dtile * 16 + low) * R_;

  v8f acc = {};
#pragma unroll
  for (int k = 0; k < R_; k += 4) {
    v2f a = *(const v2f*)(Arow + k + 2 * hi);
    v2f b = *(const v2f*)(Brow + k + 2 * hi);
    acc = __builtin_amdgcn_wmma_f32_16x16x4_f32(false, a, false, b,
                                                (short)0, acc, false, false);
  }
  const int dcol = dtile * 16 + low;
  const float bias = bdt[dcol];
  float* Out = dt + (size_t)(mtile * 16) * D_ + dcol;
#pragma unroll
  for (int r = 0; r < 8; ++r) {
    float z  = acc[r] + bias;
    // numerically stable softplus
    float sp = fmaxf(z, 0.0f) + log1pf(__expf(-fabsf(z)));
    Out[(size_t)(r + 8 * hi) * D_] = sp;
  }
}

// ---------------------------------------------------------------------------
// helpers for the scan kernels
// ---------------------------------------------------------------------------
__device__ __forceinline__ void load16(const float* __restrict__ p, float v[16]) {
  const float4* q = (const float4*)p;
  float4 a = q[0], b = q[1], c = q[2], d = q[3];
  v[0]=a.x; v[1]=a.y; v[2]=a.z; v[3]=a.w;
  v[4]=b.x; v[5]=b.y; v[6]=b.z; v[7]=b.w;
  v[8]=c.x; v[9]=c.y; v[10]=c.z; v[11]=c.w;
  v[12]=d.x; v[13]=d.y; v[14]=d.z; v[15]=d.w;
}
__device__ __forceinline__ void store16(float* __restrict__ p, const float v[16]) {
  float4* q = (float4*)p;
  q[0] = make_float4(v[0],v[1],v[2],v[3]);
  q[1] = make_float4(v[4],v[5],v[6],v[7]);
  q[2] = make_float4(v[8],v[9],v[10],v[11]);
  q[3] = make_float4(v[12],v[13],v[14],v[15]);
}

// thread index decomposition shared by pass1/pass3:
//   lanes <-> consecutive d (coalesced dt/x/out columns), waves share (b,chunk)
#define DECOMP(idx, b, d, chunk)                \
  const int dlow  = (idx) & 31;                 \
  const int dhigh = ((idx) >> 5) & 31;          \
  const int chunk = ((idx) >> 10) & 31;         \
  const int b     = (idx) >> 15;                \
  const int d     = dhigh * 32 + dlow;

// ---------------------------------------------------------------------------
// Pass 1: local scan per (b, d, chunk) from h=0; record H_last and decay
//         product P_last (per state n).  65,536 threads.
// ---------------------------------------------------------------------------
__global__ __launch_bounds__(256) void k_scan_pass1(
    const float* __restrict__ x, const float* __restrict__ xdbl,
    const float* __restrict__ dt, const float* __restrict__ A_log,
    float* __restrict__ Hlast, float* __restrict__ Plast) {
  const int idx = blockIdx.x * 256 + threadIdx.x;
  DECOMP(idx, b, d, chunk);

  float Ac[N_];
#pragma unroll
  for (int n = 0; n < N_; ++n) Ac[n] = -__expf(A_log[d * N_ + n]);

  float h[N_], p[N_];
#pragma unroll
  for (int n = 0; n < N_; ++n) { h[n] = 0.0f; p[n] = 1.0f; }

  const int l0 = chunk * TCH;
  const float* dtp = dt   + (size_t)(b * L_ + l0) * D_ + d;
  const float* xp  = x    + (size_t)(b * L_ + l0) * D_ + d;
  const float* bc  = xdbl + (size_t)(b * L_ + l0) * E_ + R_;   // B cols of x_dbl

  for (int l = 0; l < TCH; ++l) {
    const float dtv = dtp[(size_t)l * D_];
    const float xv  = xp[(size_t)l * D_];
    float Bn[N_];
    load16(bc + (size_t)l * E_, Bn);      // wave-uniform broadcast load
    const float dx = dtv * xv;
#pragma unroll
    for (int n = 0; n < N_; ++n) {
      const float dA = __expf(dtv * Ac[n]);
      h[n] = dA * h[n] + dx * Bn[n];
      p[n] *= dA;
    }
  }
  const size_t off = ((size_t)(b * CHUNKS + chunk) * D_ + d) * N_;
  store16(Hlast + off, h);
  store16(Plast + off, p);
}

// ---------------------------------------------------------------------------
// Pass 2: 32-step chunk-summary recurrence per (b,d): H_{c+1} = Hlast_c + Plast_c*H_c
//         Emits the entering state H_init for every chunk.  2,048 threads.
// ---------------------------------------------------------------------------
__global__ __launch_bounds__(256) void k_scan_pass2(
    const float* __restrict__ Hlast, const float* __restrict__ Plast,
    float* __restrict__ Hinit) {
  const int idx = blockIdx.x * 256 + threadIdx.x;   // 0..2047
  const int d = idx & (D_ - 1);
  const int b = idx >> 10;

  float H[N_];
#pragma unroll
  for (int n = 0; n < N_; ++n) H[n] = 0.0f;

  for (int c = 0; c < CHUNKS; ++c) {
    const size_t off = ((size_t)(b * CHUNKS + c) * D_ + d) * N_;
    store16(Hinit + off, H);
    float hl[N_], pl[N_];
    load16(Hlast + off, hl);
    load16(Plast + off, pl);
#pragma unroll
    for (int n = 0; n < N_; ++n) H[n] = hl[n] + pl[n] * H[n];
  }
}

// ---------------------------------------------------------------------------
// Pass 3: re-run each chunk from its true entering state, emit y + x*D.
//         65,536 threads; dt/x_dbl are L2-resident from pass 1.
// ---------------------------------------------------------------------------
__global__ __launch_bounds__(256) void k_scan_pass3(
    const float* __restrict__ x, const float* __restrict__ xdbl,
    const float* __restrict__ dt, const float* __restrict__ A_log,
    const float* __restrict__ Dparam, const float* __restrict__ Hinit,
    float* __restrict__ out) {
  const int idx = blockIdx.x * 256 + threadIdx.x;
  DECOMP(idx, b, d, chunk);

  float Ac[N_];
#pragma unroll
  for (int n = 0; n < N_; ++n) Ac[n] = -__expf(A_log[d * N_ + n]);

  float h[N_];
  load16(Hinit + ((size_t)(b * CHUNKS + chunk) * D_ + d) * N_, h);
  const float Dp = Dparam[d];

  const int l0 = chunk * TCH;
  const float* dtp = dt   + (size_t)(b * L_ + l0) * D_ + d;
  const float* xp  = x    + (size_t)(b * L_ + l0) * D_ + d;
  const float* bc  = xdbl + (size_t)(b * L_ + l0) * E_ + R_;
  float*       op  = out  + (size_t)(b * L_ + l0) * D_ + d;

  for (int l = 0; l < TCH; ++l) {
    const float dtv = dtp[(size_t)l * D_];
    const float xv  = xp[(size_t)l * D_];
    float Bn[N_], Cn[N_];
    load16(bc + (size_t)l * E_,       Bn);
    load16(bc + (size_t)l * E_ + N_,  Cn);
    const float dx = dtv * xv;
    float y = 0.0f;
#pragma unroll
    for (int n = 0; n < N_; ++n) {
      const float dA = __expf(dtv * Ac[n]);
      h[n] = dA * h[n] + dx * Bn[n];
      y += h[n] * Cn[n];
    }
    op[(size_t)l * D_] = y + xv * Dp;
  }
}

// ---------------------------------------------------------------------------
extern "C" void kernel_launch(void* const* d_in, const int* in_sizes, int n_in,
                              void* d_out, int out_size, void* d_ws, size_t ws_size,
                              hipStream_t stream) {
  const float* x      = (const float*)d_in[0];
  const float* Wx     = (const float*)d_in[1];
  const float* Wdt    = (const float*)d_in[2];
  const float* bdt    = (const float*)d_in[3];
  const float* A_log  = (const float*)d_in[4];
  const float* Dparam = (const float*)d_in[5];
  float* out = (float*)d_out;

  // workspace layout (floats): xdbl | dt | Hlast | Plast | Hinit  (~31 MB)
  float* xdbl  = (float*)d_ws;
  float* dtws  = xdbl  + (size_t)(B_ * L_) * E_;            // 4096*96
  float* Hlast = dtws  + (size_t)(B_ * L_) * D_;            // 4096*1024
  float* Plast = Hlast + (size_t)B_ * CHUNKS * D_ * N_;     // 2*32*1024*16
  float* Hinit = Plast + (size_t)B_ * CHUNKS * D_ * N_;

  // GEMM 1: 256*6 = 1536 tiles, 4 waves/block
  k_xproj<<<384, 128, 0, stream>>>(x, Wx, xdbl);
  // GEMM 2: 256*64 = 16384 tiles, 4 waves/block
  k_dtproj<<<4096, 128, 0, stream>>>(xdbl, Wdt, bdt, dtws);
  // chunked linear-recurrence scan
  k_scan_pass1<<<256, 256, 0, stream>>>(x, xdbl, dtws, A_log, Hlast, Plast);
  k_scan_pass2<<<8, 256, 0, stream>>>(Hlast, Plast, Hinit);
  k_scan_pass3<<<256, 256, 0, stream>>>(x, xdbl, dtws, A_log, Dparam, Hinit, out);
}